// TemporalGNN_69526930588439
// MI455X (gfx1250) — compile-verified
//
#include <hip/hip_runtime.h>
#include <hip/hip_bf16.h>
#include <math.h>

// ---------------- problem constants ----------------
#define NN 20000
#define DD 256
#define SS 4
#define EE 320000
#define HH 5
#define LL 3
#define EPS 1e-5f

typedef float v2f __attribute__((ext_vector_type(2)));
typedef float v8f __attribute__((ext_vector_type(8)));

// epilogue modes
#define EPI_NONE 0
#define EPI_BIAS 1
#define EPI_BIAS_GELU 2

// ---------------- WMMA f32 GEMM ----------------
// C[N,M] = A[N,K] @ B[K,M] (+ bias[M]) (+ exact GELU)
// Wave tile: 16 rows x 64 cols (4x v_wmma_f32_16x16x4_f32 accumulators)
// Block: 256 threads = 8 waves -> 128 rows x 64 cols per block.
// B panels (32 x 64) double-buffered in LDS, stored PRE-SWIZZLED as
// [q=k/4][col][half][pair] so each lane's B fragment is one aligned
// ds_load_b64; staging writes one ds_store_b128 per (q,col) quad.
// Out-of-range rows/cols are CLAMPED on load (stores are guarded), so the
// hot loop has no exec-mask predication and all addressing is 32-bit.
#define GEMM_BK 32
#define GEMM_TN 64

__global__ __launch_bounds__(256)
void wmma_gemm_f32(const float* __restrict__ A, const float* __restrict__ B,
                   const float* __restrict__ bias, float* __restrict__ C,
                   int Nrows, int K, int M, int epi)
{
    // [buf][q][col][half][pair] : 2 * 8 * 64 * 2 * 2 floats = 16 KB
    __shared__ __align__(16) float Bsw[2][GEMM_BK / 4][GEMM_TN][2][2];

    const int lane = threadIdx.x & 31;
    const int wave = threadIdx.x >> 5;
    const int half = lane >> 4;      // 0: K pair {0,1}, 1: K pair {2,3}
    const int mrow = lane & 15;      // row within 16-row tile / col within 16-col tile

    const int row0 = blockIdx.x * 128 + wave * 16;
    const int col0 = blockIdx.y * GEMM_TN;

    // clamped A row (stores remain guarded, duplicate rows are discarded)
    int arow = row0 + mrow;
    if (arow >= Nrows) arow = Nrows - 1;
    const float* __restrict__ Arow = A + (size_t)arow * K + half * 2;

    // B staging: 512 (q,col) quads, 2 per thread; clamped column, 32-bit offsets
    const int nPanels = K / GEMM_BK;

    auto stage = [&](int buf, int k0) {
        #pragma unroll
        for (int j = 0; j < 2; ++j) {
            int pairIdx = threadIdx.x + j * 256;      // 0..511
            int q  = pairIdx >> 6;                    // 0..7
            int cc = pairIdx & 63;
            int colc = col0 + cc;
            if (colc >= M) colc = M - 1;              // clamp, no predication
            int base = (k0 + q * 4) * M + colc;       // fits in 32-bit
            float4 v;
            v.x = B[base];
            v.y = B[base + M];
            v.z = B[base + 2 * M];
            v.w = B[base + 3 * M];
            *(float4*)&Bsw[buf][q][cc][0][0] = v;     // ds_store_b128
        }
    };

    v8f acc[4] = {};

    stage(0, 0);
    __syncthreads();

    for (int pnl = 0; pnl < nPanels; ++pnl) {
        const int buf = pnl & 1;
        if (pnl + 1 < nPanels) stage(buf ^ 1, (pnl + 1) * GEMM_BK);

        const float* __restrict__ Ap = Arow + pnl * GEMM_BK;
        #pragma unroll
        for (int q = 0; q < 8; ++q) {
            // A fragment: lane holds A[row][k0+4q + half*2 + {0,1}]
            const v2f a = *(const v2f*)(Ap + q * 4);
            #pragma unroll
            for (int t = 0; t < 4; ++t) {
                // one aligned 8B LDS read: {B[4q+2h][c], B[4q+2h+1][c]}
                const v2f b = *(const v2f*)&Bsw[buf][q][t * 16 + mrow][half][0];
                acc[t] = __builtin_amdgcn_wmma_f32_16x16x4_f32(
                    /*neg_a=*/false, a, /*neg_b=*/false, b,
                    /*c_mod=*/(short)0, acc[t],
                    /*reuse_a=*/false, /*reuse_b=*/false);
            }
        }
        __syncthreads();
    }

    // epilogue + store (C layout: VGPR r -> row r (lanes 0-15) / r+8 (lanes 16-31))
    #pragma unroll
    for (int t = 0; t < 4; ++t) {
        int col = col0 + t * 16 + mrow;
        if (col >= M) continue;
        float bv = (epi != EPI_NONE) ? bias[col] : 0.0f;
        #pragma unroll
        for (int r = 0; r < 8; ++r) {
            int row = row0 + r + half * 8;
            if (row >= Nrows) continue;
            float v = acc[t][r] + bv;
            if (epi == EPI_BIAS_GELU)
                v = 0.5f * v * (1.0f + erff(v * 0.70710678118654752f));
            C[(size_t)row * M + col] = v;
        }
    }
}

// ---------------- graph preprocessing ----------------
__global__ void fill_val(float* p, float v, size_t n) {
    size_t i = (size_t)blockIdx.x * blockDim.x + threadIdx.x;
    if (i < n) p[i] = v;
}

__global__ void deg_scatter(const int* __restrict__ dst, const float* __restrict__ w,
                            float* __restrict__ deg, int E) {
    int e = blockIdx.x * blockDim.x + threadIdx.x;
    if (e < E) atomicAdd(&deg[dst[e]], w[e]);
}

__global__ void dinv_kernel(const float* __restrict__ deg, float* __restrict__ dinv, int n) {
    int i = blockIdx.x * blockDim.x + threadIdx.x;
    if (i >= n) return;
    float d = deg[i];
    dinv[i] = (d > 0.0f) ? rsqrtf(fmaxf(d, EPS)) : 0.0f;
}

// ---------------- GCN aggregation ----------------
// self-loop (norm = dinv^2) + bias, initializes the accumulator
__global__ void gcn_self_bias(const float* __restrict__ hW, const float* __restrict__ dinv,
                              const float* __restrict__ bias, float* __restrict__ out,
                              int Nr, int D) {
    size_t i = (size_t)blockIdx.x * blockDim.x + threadIdx.x;
    if (i >= (size_t)Nr * D) return;
    int n = (int)(i / D);
    int d = (int)(i % D);
    float di = dinv[n];
    out[i] = bias[d] + di * di * hW[i];
}

// per edge: out[dst] += dinv[src]*w*dinv[dst] * hW[src]; 64 lanes/edge, float4 gather
__global__ void gcn_scatter(const float* __restrict__ hW, const int* __restrict__ src,
                            const int* __restrict__ dst, const float* __restrict__ w,
                            const float* __restrict__ dinv, float* __restrict__ out,
                            int E, int D) {
    size_t idx = (size_t)blockIdx.x * blockDim.x + threadIdx.x;
    int e = (int)(idx >> 6);
    int c = (int)((idx & 63) << 2);
    if (e >= E) return;
    int s = src[e];
    int t = dst[e];
    float nrm = dinv[s] * w[e] * dinv[t];
    const float4 v = *(const float4*)(hW + (size_t)s * D + c);
    float* o = out + (size_t)t * D + c;
    atomicAdd(o + 0, nrm * v.x);
    atomicAdd(o + 1, nrm * v.y);
    atomicAdd(o + 2, nrm * v.z);
    atomicAdd(o + 3, nrm * v.w);
}

// ---------------- LayerNorm + ReLU (one wave32 per 256-wide row) ----------------
__global__ __launch_bounds__(256)
void layernorm_relu(const float* __restrict__ in, const float* __restrict__ g,
                    const float* __restrict__ b, float* __restrict__ out, int Nr) {
    const int lane = threadIdx.x & 31;
    const int wave = threadIdx.x >> 5;
    const int row = blockIdx.x * 8 + wave;
    if (row >= Nr) return;
    const float* x = in + (size_t)row * DD;

    float vals[8];
    float s = 0.0f;
    #pragma unroll
    for (int i = 0; i < 8; ++i) { vals[i] = x[lane + i * 32]; s += vals[i]; }
    #pragma unroll
    for (int off = 16; off > 0; off >>= 1) s += __shfl_xor(s, off, 32);
    float mean = s * (1.0f / DD);

    float vs = 0.0f;
    #pragma unroll
    for (int i = 0; i < 8; ++i) { float d = vals[i] - mean; vs += d * d; }
    #pragma unroll
    for (int off = 16; off > 0; off >>= 1) vs += __shfl_xor(vs, off, 32);
    float rstd = rsqrtf(vs * (1.0f / DD) + EPS);

    float* o = out + (size_t)row * DD;
    #pragma unroll
    for (int i = 0; i < 8; ++i) {
        int col = lane + i * 32;
        float y = (vals[i] - mean) * rstd * g[col] + b[col];
        o[col] = fmaxf(y, 0.0f);
    }
}

// ---------------- GRU gate fusion (in-place h update) ----------------
__global__ void gru_gate(const float* __restrict__ gi, const float* __restrict__ gh,
                         float* __restrict__ h, int Nr, int D) {
    size_t i = (size_t)blockIdx.x * blockDim.x + threadIdx.x;
    if (i >= (size_t)Nr * D) return;
    int n = (int)(i / D);
    int d = (int)(i % D);
    const float* gin = gi + (size_t)n * 3 * D;
    const float* ghn = gh + (size_t)n * 3 * D;
    float r  = 1.0f / (1.0f + expf(-(gin[d] + ghn[d])));
    float z  = 1.0f / (1.0f + expf(-(gin[D + d] + ghn[D + d])));
    float ng = tanhf(gin[2 * D + d] + r * ghn[2 * D + d]);
    h[i] = (1.0f - z) * ng + z * h[i];
}

// ---------------- weight transpose (M x K -> K x M) ----------------
__global__ void transpose2d(const float* __restrict__ in, float* __restrict__ out,
                            int R, int C) {
    size_t i = (size_t)blockIdx.x * blockDim.x + threadIdx.x;
    if (i >= (size_t)R * C) return;
    int r = (int)(i / C);
    int c = (int)(i % C);
    out[(size_t)c * R + r] = in[i];
}

// ---------------- host orchestration ----------------
static inline void launch_gemm(const float* A, const float* B, const float* bias,
                               float* C, int Nr, int K, int M, int epi,
                               hipStream_t s) {
    dim3 grid((Nr + 127) / 128, (M + GEMM_TN - 1) / GEMM_TN);
    wmma_gemm_f32<<<grid, 256, 0, s>>>(A, B, bias, C, Nr, K, M, epi);
}

extern "C" void kernel_launch(void* const* d_in, const int* in_sizes, int n_in,
                              void* d_out, int out_size, void* d_ws, size_t ws_size,
                              hipStream_t stream) {
    const float* x      = (const float*)d_in[0];   // [S,N,D]
    const int*   esrc   = (const int*)d_in[1];
    const int*   edst   = (const int*)d_in[2];
    const float* ew     = (const float*)d_in[3];
    const float* gcn_W  = (const float*)d_in[4];   // [L,D,D]
    const float* gcn_b  = (const float*)d_in[5];   // [L,D]
    const float* ln_g   = (const float*)d_in[6];
    const float* ln_b   = (const float*)d_in[7];
    const float* Wih    = (const float*)d_in[8];   // [3D,D]
    const float* Whh    = (const float*)d_in[9];
    const float* bih    = (const float*)d_in[10];  // [3D]
    const float* bhh    = (const float*)d_in[11];
    const float* fc1_W  = (const float*)d_in[12];  const float* fc1_b = (const float*)d_in[13];
    const float* fc2_W  = (const float*)d_in[14];  const float* fc2_b = (const float*)d_in[15];
    const float* fc3_W  = (const float*)d_in[16];  const float* fc3_b = (const float*)d_in[17];
    const float* dh1_W  = (const float*)d_in[18];  const float* dh1_b = (const float*)d_in[19];
    const float* dh2_W  = (const float*)d_in[20];  const float* dh2_b = (const float*)d_in[21];

    float* out = (float*)d_out;  // forecast [N,H] then direction [N,H]

    // ---- workspace carve-up (floats) ----
    float* ws = (float*)d_ws;
    size_t off = 0;
    float* deg    = ws + off; off += NN;
    float* dinv   = ws + off; off += NN;
    float* hbuf   = ws + off; off += (size_t)NN * DD;
    float* hW     = ws + off; off += (size_t)NN * DD;
    float* agg    = ws + off; off += (size_t)NN * DD;
    float* hstate = ws + off; off += (size_t)NN * DD;
    float* gi     = ws + off; off += (size_t)NN * 3 * DD;
    float* gh     = ws + off; off += (size_t)NN * 3 * DD;
    float* WihT   = ws + off; off += (size_t)3 * DD * DD;
    float* WhhT   = ws + off; off += (size_t)3 * DD * DD;
    float* fc1T   = ws + off; off += (size_t)DD * (DD / 2);
    float* fc2T   = ws + off; off += (size_t)(DD / 2) * (DD / 4);
    float* fc3T   = ws + off; off += (size_t)(DD / 4) * HH;
    float* dh1T   = ws + off; off += (size_t)DD * (DD / 4);
    float* dh2T   = ws + off; off += (size_t)(DD / 4) * HH;
    float* f1     = ws + off; off += (size_t)NN * (DD / 2);
    float* f2     = ws + off; off += (size_t)NN * (DD / 4);
    float* db1    = ws + off; off += (size_t)NN * (DD / 4);

    const int T = 256;
    auto blocks = [](size_t n, int t) { return (unsigned)((n + t - 1) / t); };

    // ---- 1. degree / dinv (self-loop weight folded into init of 1.0) ----
    fill_val<<<blocks(NN, T), T, 0, stream>>>(deg, 1.0f, NN);
    deg_scatter<<<blocks(EE, T), T, 0, stream>>>(edst, ew, deg, EE);
    dinv_kernel<<<blocks(NN, T), T, 0, stream>>>(deg, dinv, NN);

    // ---- 2. transpose weights used as A @ W^T ----
    transpose2d<<<blocks((size_t)3 * DD * DD, T), T, 0, stream>>>(Wih, WihT, 3 * DD, DD);
    transpose2d<<<blocks((size_t)3 * DD * DD, T), T, 0, stream>>>(Whh, WhhT, 3 * DD, DD);
    transpose2d<<<blocks((size_t)(DD / 2) * DD, T), T, 0, stream>>>(fc1_W, fc1T, DD / 2, DD);
    transpose2d<<<blocks((size_t)(DD / 4) * (DD / 2), T), T, 0, stream>>>(fc2_W, fc2T, DD / 4, DD / 2);
    transpose2d<<<blocks((size_t)HH * (DD / 4), T), T, 0, stream>>>(fc3_W, fc3T, HH, DD / 4);
    transpose2d<<<blocks((size_t)(DD / 4) * DD, T), T, 0, stream>>>(dh1_W, dh1T, DD / 4, DD);
    transpose2d<<<blocks((size_t)HH * (DD / 4), T), T, 0, stream>>>(dh2_W, dh2T, HH, DD / 4);

    // ---- 3. GRU state = 0 ----
    fill_val<<<blocks((size_t)NN * DD, T), T, 0, stream>>>(hstate, 0.0f, (size_t)NN * DD);

    // ---- 4. snapshots: 3x(GCN + LN + ReLU) then one GRU step each ----
    for (int si = 0; si < SS; ++si) {
        for (int l = 0; l < LL; ++l) {
            const float* Ain = (l == 0) ? (x + (size_t)si * NN * DD) : hbuf;
            // hW = Ain @ gcn_W[l]
            launch_gemm(Ain, gcn_W + (size_t)l * DD * DD, nullptr, hW,
                        NN, DD, DD, EPI_NONE, stream);
            // agg = bias + dinv^2 * hW  (self loops), then edge scatter
            gcn_self_bias<<<blocks((size_t)NN * DD, T), T, 0, stream>>>(
                hW, dinv, gcn_b + (size_t)l * DD, agg, NN, DD);
            gcn_scatter<<<blocks((size_t)EE * 64, T), T, 0, stream>>>(
                hW, esrc, edst, ew, dinv, agg, EE, DD);
            // hbuf = relu(LN(agg))
            layernorm_relu<<<blocks((size_t)NN, 8), 256, 0, stream>>>(
                agg, ln_g + (size_t)l * DD, ln_b + (size_t)l * DD, hbuf, NN);
        }
        // GRU step: gi = hbuf @ Wih^T + bih ; gh = hstate @ Whh^T + bhh
        launch_gemm(hbuf,   WihT, bih, gi, NN, DD, 3 * DD, EPI_BIAS, stream);
        launch_gemm(hstate, WhhT, bhh, gh, NN, DD, 3 * DD, EPI_BIAS, stream);
        gru_gate<<<blocks((size_t)NN * DD, T), T, 0, stream>>>(gi, gh, hstate, NN, DD);
    }

    // ---- 5. heads ----
    launch_gemm(hstate, fc1T, fc1_b, f1, NN, DD, DD / 2, EPI_BIAS_GELU, stream);
    launch_gemm(f1, fc2T, fc2_b, f2, NN, DD / 2, DD / 4, EPI_BIAS_GELU, stream);
    launch_gemm(f2, fc3T, fc3_b, out, NN, DD / 4, HH, EPI_BIAS, stream);  // forecast

    launch_gemm(hstate, dh1T, dh1_b, db1, NN, DD, DD / 4, EPI_BIAS_GELU, stream);
    launch_gemm(db1, dh2T, dh2_b, out + (size_t)NN * HH, NN, DD / 4, HH, EPI_BIAS, stream); // direction

    (void)in_sizes; (void)n_in; (void)out_size; (void)ws_size;
}